// AttentionAggregator_61795989454872
// MI455X (gfx1250) — compile-verified
//
#include <hip/hip_runtime.h>
#include <hip/hip_bf16.h>

// ---------------------------------------------------------------------------
// GAT-style attention aggregator for MI455X (gfx1250, wave32).
//
//   new_emb = feature_b @ W + bias            -> FP32 WMMA 16x16x4 GEMM
//   fa_proj = feature_a @ a1 ; nb_proj = new_emb @ a2
//   per edge: score = exp(elu(fa_proj[src] + nb_proj[dst], 0.1))
//             num[src] += score * new_emb[dst] ; den[src] += score
//   out = num / den (den==0 -> 1)
//
// Workspace layout (floats):
//   new_emb : n_b*64
//   fa_proj : n_a
//   nb_proj : n_b
//   den     : n_a
// d_out is used directly as the `num` accumulator, then divided in place.
// ---------------------------------------------------------------------------

typedef __attribute__((ext_vector_type(2))) float v2f;
typedef __attribute__((ext_vector_type(8))) float v8f;

#define DIM 64

// ---------------- zero accumulators (out + den) -----------------------------
__global__ void aa_zero_kernel(float* __restrict__ out, long n_out,
                               float* __restrict__ den, long n_den) {
    long i = (long)blockIdx.x * blockDim.x + threadIdx.x;
    if (i < n_out) out[i] = 0.0f;
    if (i < n_den) den[i] = 0.0f;
}

// ---------------- FP32 WMMA GEMM: new_emb = feature_b @ W + bias ------------
// One wave per 16-row tile of feature_b. N = K = 64.
// A (16x4 f32, 2 VGPRs): lane%16 = M, lane/16 selects K-pair, vgpr = K lsb.
// B (4x16 f32, 2 VGPRs): K = vgpr + 2*(lane/16), N = lane%16.
// C (16x16 f32, 8 VGPRs): M = vgpr + 8*(lane/16), N = lane%16.
__global__ void aa_gemm_wmma_kernel(const float* __restrict__ fb,
                                    const float* __restrict__ W,
                                    const float* __restrict__ bias,
                                    float* __restrict__ new_emb,
                                    int n_b) {
    const int lane  = threadIdx.x & 31;
    const int wave  = threadIdx.x >> 5;
    const int mtile = blockIdx.x * (blockDim.x >> 5) + wave;
    const int m0    = mtile * 16;
    if (m0 >= n_b) return;                 // wave-uniform: EXEC stays all-1s

    const int half = lane >> 4;            // 0 for lanes 0-15, 1 for 16-31
    const int l16  = lane & 15;

    v8f c0 = {}, c1 = {}, c2 = {}, c3 = {};

    int arow = m0 + l16;
    if (arow >= n_b) arow = n_b - 1;       // clamp (stores are guarded later)
    const float* __restrict__ ap = fb + (long)arow * DIM + half * 2;

    #pragma unroll 4
    for (int kk = 0; kk < DIM; kk += 4) {
        v2f a = *(const v2f*)(ap + kk);

        const int   kb = kk + half * 2;
        const float* w0 = W + (long)kb * DIM + l16;       // K = kb
        const float* w1 = w0 + DIM;                        // K = kb + 1

        v2f b0 = { w0[ 0], w1[ 0] };
        v2f b1 = { w0[16], w1[16] };
        v2f b2 = { w0[32], w1[32] };
        v2f b3 = { w0[48], w1[48] };

        c0 = __builtin_amdgcn_wmma_f32_16x16x4_f32(false, a, false, b0, (short)0, c0, false, false);
        c1 = __builtin_amdgcn_wmma_f32_16x16x4_f32(false, a, false, b1, (short)0, c1, false, false);
        c2 = __builtin_amdgcn_wmma_f32_16x16x4_f32(false, a, false, b2, (short)0, c2, false, false);
        c3 = __builtin_amdgcn_wmma_f32_16x16x4_f32(false, a, false, b3, (short)0, c3, false, false);
    }

    const float bi0 = bias[l16 +  0];
    const float bi1 = bias[l16 + 16];
    const float bi2 = bias[l16 + 32];
    const float bi3 = bias[l16 + 48];

    const int mbase = m0 + half * 8;
    #pragma unroll
    for (int r = 0; r < 8; ++r) {
        const int row = mbase + r;
        if (row < n_b) {
            float* __restrict__ o = new_emb + (long)row * DIM + l16;
            o[ 0] = c0[r] + bi0;
            o[16] = c1[r] + bi1;
            o[32] = c2[r] + bi2;
            o[48] = c3[r] + bi3;
        }
    }
}

// ---------------- row projections: fa_proj, nb_proj -------------------------
// One wave per row; lane handles 2 columns; shfl-xor wave reduction.
__global__ void aa_proj_kernel(const float* __restrict__ fa,
                               const float* __restrict__ new_emb,
                               const float* __restrict__ a_vec,
                               float* __restrict__ fa_proj,
                               float* __restrict__ nb_proj,
                               int n_a, int n_b) {
    const int lane = threadIdx.x & 31;
    const long row = (long)blockIdx.x * (blockDim.x >> 5) + (threadIdx.x >> 5);

    const float* src;
    const float* av;
    float*       dst;
    if (row < n_a) {
        src = fa + row * DIM;      av = a_vec;      dst = fa_proj + row;
    } else if (row < (long)n_a + n_b) {
        const long r = row - n_a;
        src = new_emb + r * DIM;   av = a_vec + 64; dst = nb_proj + r;
    } else {
        return;
    }

    float2 x = *(const float2*)(src + lane * 2);
    float2 w = *(const float2*)(av  + lane * 2);
    float s = x.x * w.x + x.y * w.y;
    #pragma unroll
    for (int off = 16; off > 0; off >>= 1)
        s += __shfl_xor(s, off, 32);
    if (lane == 0) *dst = s;
}

// ---------------- edge kernel: score + scatter-add --------------------------
// One wave per edge; lane covers columns [2*lane, 2*lane+1].
__global__ void aa_edge_kernel(const int* __restrict__ edges,
                               const float* __restrict__ fa_proj,
                               const float* __restrict__ nb_proj,
                               const float* __restrict__ new_emb,
                               float* __restrict__ num,
                               float* __restrict__ den,
                               long n_e) {
    const int  lane = threadIdx.x & 31;
    const long e    = (long)blockIdx.x * (blockDim.x >> 5) + (threadIdx.x >> 5);
    if (e >= n_e) return;

    const int src = edges[e * 2 + 0];
    const int dst = edges[e * 2 + 1];

    const float s     = fa_proj[src] + nb_proj[dst];
    const float elu   = (s > 0.0f) ? s : 0.1f * (__expf(s) - 1.0f);
    const float score = __expf(elu);

    const float2 nb = *(const float2*)(new_emb + (long)dst * DIM + lane * 2);
    float* __restrict__ o = num + (long)src * DIM + lane * 2;
    atomicAdd(o + 0, score * nb.x);
    atomicAdd(o + 1, score * nb.y);
    if (lane == 0) atomicAdd(den + src, score);
}

// ---------------- finalize: out = num / den ---------------------------------
__global__ void aa_finalize_kernel(float* __restrict__ out,
                                   const float* __restrict__ den,
                                   long n_elems) {
    const long i = (long)blockIdx.x * blockDim.x + threadIdx.x;
    if (i >= n_elems) return;
    float d = den[i >> 6];
    d = (d == 0.0f) ? 1.0f : d;
    out[i] = out[i] / d;
}

// ---------------------------------------------------------------------------
extern "C" void kernel_launch(void* const* d_in, const int* in_sizes, int n_in,
                              void* d_out, int out_size, void* d_ws, size_t ws_size,
                              hipStream_t stream) {
    const float* fa    = (const float*)d_in[0];   // feature_a (n_a,64)
    const float* fb    = (const float*)d_in[1];   // feature_b (n_b,64)
    const float* W     = (const float*)d_in[2];   // (64,64)
    const float* bias  = (const float*)d_in[3];   // (64,)
    const float* a_vec = (const float*)d_in[4];   // (128,1)
    const int*   edges = (const int*)d_in[5];     // (E,2)

    const int  n_a = in_sizes[0] / DIM;
    const int  n_b = in_sizes[1] / DIM;
    const long n_e = (long)in_sizes[5] / 2;

    float* out = (float*)d_out;

    float* ws      = (float*)d_ws;
    float* new_emb = ws;                              // n_b*64
    float* fa_proj = new_emb + (long)n_b * DIM;       // n_a
    float* nb_proj = fa_proj + n_a;                   // n_b
    float* den     = nb_proj + n_b;                   // n_a

    // 1) zero num (d_out) and den
    {
        const long n_out = (long)n_a * DIM;
        const long n_den = n_a;
        const long n_max = n_out > n_den ? n_out : n_den;
        const int  blocks = (int)((n_max + 255) / 256);
        aa_zero_kernel<<<blocks, 256, 0, stream>>>(out, n_out, den, n_den);
    }

    // 2) WMMA GEMM: new_emb = fb @ W + bias   (4 waves / block, 1 m-tile / wave)
    {
        const int mtiles = (n_b + 15) / 16;
        const int blocks = (mtiles + 3) / 4;
        aa_gemm_wmma_kernel<<<blocks, 128, 0, stream>>>(fb, W, bias, new_emb, n_b);
    }

    // 3) projections (8 waves / block, 1 row / wave)
    {
        const long rows   = (long)n_a + n_b;
        const int  blocks = (int)((rows + 7) / 8);
        aa_proj_kernel<<<blocks, 256, 0, stream>>>(fa, new_emb, a_vec,
                                                   fa_proj, nb_proj, n_a, n_b);
    }

    // 4) edges (8 waves / block, 1 edge / wave)
    {
        const int blocks = (int)((n_e + 7) / 8);
        aa_edge_kernel<<<blocks, 256, 0, stream>>>(edges, fa_proj, nb_proj,
                                                   new_emb, out, den, n_e);
    }

    // 5) finalize out = num / den
    {
        const long n_elems = (long)n_a * DIM;
        const int  blocks  = (int)((n_elems + 255) / 256);
        aa_finalize_kernel<<<blocks, 256, 0, stream>>>(out, den, n_elems);
    }
}